// MultiHeadAttention_69398081569166
// MI455X (gfx1250) — compile-verified
//
#include <hip/hip_runtime.h>
#include <math.h>

// Problem constants (match reference)
#define BB 2
#define SS 2048
#define DD 1024
#define HH 16
#define DKK 64
#define RMS_EPS 1.1920929e-07f

typedef float v2f __attribute__((ext_vector_type(2)));
typedef float v8f __attribute__((ext_vector_type(8)));

__device__ __forceinline__ v8f wmma4(v2f a, v2f b, v8f c) {
  // V_WMMA_F32_16X16X4_F32 : D(16x16 f32) = A(16x4) * B(4x16) + C
  return __builtin_amdgcn_wmma_f32_16x16x4_f32(
      /*neg_a=*/false, a, /*neg_b=*/false, b,
      /*c_mod=*/(short)0, c, /*reuse_a=*/false, /*reuse_b=*/false);
}

// ---------------------------------------------------------------------------
// 1) RMSNorm: normed[row,d] = x[row,d] * rsqrt(mean(x^2)+eps) * g[d]
// ---------------------------------------------------------------------------
__global__ __launch_bounds__(256) void rmsnorm_kernel(const float* __restrict__ x,
                                                      const float* __restrict__ g,
                                                      float* __restrict__ normed) {
  const int row = blockIdx.x;
  const int t = threadIdx.x;
  const float* xr = x + (size_t)row * DD;
  __shared__ float sm[256];
  float s = 0.f;
  #pragma unroll
  for (int i = 0; i < DD / 256; ++i) {
    float v = xr[t + i * 256];
    s += v * v;
  }
  sm[t] = s;
  __syncthreads();
  for (int w = 128; w > 0; w >>= 1) {
    if (t < w) sm[t] += sm[t + w];
    __syncthreads();
  }
  const float rr = rsqrtf(sm[0] / (float)DD + RMS_EPS);
  float* nr = normed + (size_t)row * DD;
  #pragma unroll
  for (int i = 0; i < DD / 256; ++i) {
    int d = t + i * 256;
    nr[d] = xr[d] * rr * g[d];
  }
}

// ---------------------------------------------------------------------------
// 2) WMMA GEMM, 32x64 macro-tile per wave (2x4 accumulators = 64 VGPRs):
//    Y[m,n] = sum_k A[m,k] * W[n,k] + bias[n] (+ res[m,n])
//    A: MxK row-major, W: NxK row-major (torch Linear weight), Y: MxN
//    Per k-step: 6 fragment loads feed 8 WMMAs; fits register budget (no spills).
// ---------------------------------------------------------------------------
__global__ __launch_bounds__(256) void gemm_wt32x64_kernel(
    const float* __restrict__ A, const float* __restrict__ W,
    const float* __restrict__ bias, const float* __restrict__ res,
    float* __restrict__ Y, int M, int N, int K) {
  const int tilesN = N >> 6;                 // 64-wide
  const int total = (M >> 5) * tilesN;       // 32-tall
  const int tile = blockIdx.x * (blockDim.x >> 5) + (threadIdx.x >> 5);
  if (tile >= total) return;                 // wave-uniform
  const int tm = tile / tilesN;
  const int tn = tile - tm * tilesN;
  const int lane = threadIdx.x & 31;
  const int hi = lane >> 4;                  // 0: K pair {0,1}, 1: K pair {2,3}
  const int lr = lane & 15;

  const float* arow0 = A + (size_t)(tm * 32 + lr) * K + 2 * hi;
  const float* arow1 = arow0 + (size_t)16 * K;
  const float* wrow0 = W + (size_t)(tn * 64 + lr) * K + 2 * hi;
  const float* wrow1 = wrow0 + (size_t)16 * K;
  const float* wrow2 = wrow0 + (size_t)32 * K;
  const float* wrow3 = wrow0 + (size_t)48 * K;

  v8f acc[8];
  #pragma unroll
  for (int i = 0; i < 8; ++i) acc[i] = (v8f){0.f,0.f,0.f,0.f,0.f,0.f,0.f,0.f};

  for (int k = 0; k < K; k += 4) {
    v2f a0, a1, b0, b1, b2, b3;
    a0.x = arow0[k]; a0.y = arow0[k + 1];
    a1.x = arow1[k]; a1.y = arow1[k + 1];
    b0.x = wrow0[k]; b0.y = wrow0[k + 1];
    b1.x = wrow1[k]; b1.y = wrow1[k + 1];
    b2.x = wrow2[k]; b2.y = wrow2[k + 1];
    b3.x = wrow3[k]; b3.y = wrow3[k + 1];
    acc[0] = wmma4(a0, b0, acc[0]);
    acc[1] = wmma4(a0, b1, acc[1]);
    acc[2] = wmma4(a0, b2, acc[2]);
    acc[3] = wmma4(a0, b3, acc[3]);
    acc[4] = wmma4(a1, b0, acc[4]);
    acc[5] = wmma4(a1, b1, acc[5]);
    acc[6] = wmma4(a1, b2, acc[6]);
    acc[7] = wmma4(a1, b3, acc[7]);
  }

  float bn[4];
  #pragma unroll
  for (int j = 0; j < 4; ++j) bn[j] = bias ? bias[tn * 64 + j * 16 + lr] : 0.f;

  #pragma unroll
  for (int i = 0; i < 2; ++i) {
    #pragma unroll
    for (int r = 0; r < 8; ++r) {
      const int m = tm * 32 + i * 16 + r + 8 * hi;
      #pragma unroll
      for (int j = 0; j < 4; ++j) {
        const int n = tn * 64 + j * 16 + lr;
        const size_t idx = (size_t)m * N + n;
        float v = acc[i * 4 + j][r] + bn[j];
        if (res) v += res[idx];
        Y[idx] = v;
      }
    }
  }
}

// ---------------------------------------------------------------------------
// 3) Scores: Sc[bh,q,k] = (Q . K)/8, causal -> -1e9.  Q,K in [B,S,D] layout.
//    16(q) x 64(k) strip per wave: Q fragment reused across 4 K-tiles.
//    Strips fully above the diagonal skip the GEMM entirely.
// ---------------------------------------------------------------------------
__global__ __launch_bounds__(256) void scores_kernel(const float* __restrict__ Q,
                                                     const float* __restrict__ Kb,
                                                     float* __restrict__ Sc) {
  const int TQ = SS >> 4;                    // 128 q-tiles
  const int TKS = TQ >> 2;                   // 32 k-strips of 64
  const int total = BB * HH * TQ * TKS;
  const int tile = blockIdx.x * (blockDim.x >> 5) + (threadIdx.x >> 5);
  if (tile >= total) return;                 // wave-uniform
  const int bh = tile / (TQ * TKS);
  const int rem = tile - bh * (TQ * TKS);
  const int qt = rem / TKS;
  const int ks = rem - qt * TKS;
  const int b = bh / HH;
  const int h = bh - b * HH;

  const int lane = threadIdx.x & 31;
  const int hi = lane >> 4;
  const int lr = lane & 15;
  const int kbase = ks * 64;

  float* srow0 = Sc + (size_t)bh * SS * SS;

  if (kbase > qt * 16 + 15) {                // strip entirely above diagonal
    #pragma unroll
    for (int r = 0; r < 8; ++r) {
      const int q = qt * 16 + r + 8 * hi;
      #pragma unroll
      for (int j = 0; j < 4; ++j)
        srow0[(size_t)q * SS + kbase + j * 16 + lr] = -1e9f;
    }
    return;
  }

  const float* qrow = Q + ((size_t)(b * SS + qt * 16 + lr)) * DD + h * DKK + 2 * hi;
  const float* krow0 = Kb + ((size_t)(b * SS + kbase + lr)) * DD + h * DKK + 2 * hi;
  const float* krow1 = krow0 + (size_t)16 * DD;
  const float* krow2 = krow0 + (size_t)32 * DD;
  const float* krow3 = krow0 + (size_t)48 * DD;

  v8f acc0 = (v8f){0.f,0.f,0.f,0.f,0.f,0.f,0.f,0.f};
  v8f acc1 = acc0, acc2 = acc0, acc3 = acc0;

  #pragma unroll
  for (int k = 0; k < DKK; k += 4) {
    v2f af, b0, b1, b2, b3;
    af.x = qrow[k]; af.y = qrow[k + 1];
    b0.x = krow0[k]; b0.y = krow0[k + 1];
    b1.x = krow1[k]; b1.y = krow1[k + 1];
    b2.x = krow2[k]; b2.y = krow2[k + 1];
    b3.x = krow3[k]; b3.y = krow3[k + 1];
    acc0 = wmma4(af, b0, acc0);
    acc1 = wmma4(af, b1, acc1);
    acc2 = wmma4(af, b2, acc2);
    acc3 = wmma4(af, b3, acc3);
  }

  #pragma unroll
  for (int r = 0; r < 8; ++r) {
    const int q = qt * 16 + r + 8 * hi;
    float vals[4] = {acc0[r], acc1[r], acc2[r], acc3[r]};
    #pragma unroll
    for (int j = 0; j < 4; ++j) {
      const int n = kbase + j * 16 + lr;
      srow0[(size_t)q * SS + n] = (n <= q) ? vals[j] * 0.125f : -1e9f;
    }
  }
}

// ---------------------------------------------------------------------------
// 4) Row softmax over S=2048 in place (rows = B*H*S)
// ---------------------------------------------------------------------------
__global__ __launch_bounds__(256) void softmax_kernel(float* __restrict__ attn) {
  const int row = blockIdx.x;
  const int t = threadIdx.x;
  float* p = attn + (size_t)row * SS;
  __shared__ float sm[256];

  float v[SS / 256];
  float mx = -INFINITY;
  #pragma unroll
  for (int i = 0; i < SS / 256; ++i) {
    v[i] = p[t + i * 256];
    mx = fmaxf(mx, v[i]);
  }
  sm[t] = mx;
  __syncthreads();
  for (int w = 128; w > 0; w >>= 1) {
    if (t < w) sm[t] = fmaxf(sm[t], sm[t + w]);
    __syncthreads();
  }
  mx = sm[0];
  __syncthreads();

  float s = 0.f;
  #pragma unroll
  for (int i = 0; i < SS / 256; ++i) {
    v[i] = expf(v[i] - mx);
    s += v[i];
  }
  sm[t] = s;
  __syncthreads();
  for (int w = 128; w > 0; w >>= 1) {
    if (t < w) sm[t] += sm[t + w];
    __syncthreads();
  }
  const float inv = 1.f / sm[0];
  #pragma unroll
  for (int i = 0; i < SS / 256; ++i) p[t + i * 256] = v[i] * inv;
}

// ---------------------------------------------------------------------------
// 5) ctx = attn @ V per (b,h): 16(q) x 64(=DK) strip per wave.
//    The HBM-resident attn fragment is loaded once, reused for 4 V-tiles.
//    k-loop stops at the diagonal (attn[q,k]==0 for k>q).
// ---------------------------------------------------------------------------
__global__ __launch_bounds__(256) void pv_kernel(const float* __restrict__ attn,
                                                 const float* __restrict__ V,
                                                 float* __restrict__ ctx) {
  const int TQ = SS >> 4;                    // 128
  const int total = BB * HH * TQ;
  const int tile = blockIdx.x * (blockDim.x >> 5) + (threadIdx.x >> 5);
  if (tile >= total) return;                 // wave-uniform
  const int bh = tile / TQ;
  const int qt = tile - bh * TQ;
  const int b = bh / HH;
  const int h = bh - b * HH;

  const int lane = threadIdx.x & 31;
  const int hi = lane >> 4;
  const int lr = lane & 15;

  const float* arow = attn + (size_t)bh * SS * SS + (size_t)(qt * 16 + lr) * SS + 2 * hi;
  const float* vb = V + (size_t)(b * SS) * DD + h * DKK + lr + (size_t)(2 * hi) * DD;

  v8f acc0 = (v8f){0.f,0.f,0.f,0.f,0.f,0.f,0.f,0.f};
  v8f acc1 = acc0, acc2 = acc0, acc3 = acc0;

  const int kmax = (qt + 1) * 16;            // causal truncation
  for (int k = 0; k < kmax; k += 4) {
    v2f af, b0, b1, b2, b3;
    af.x = arow[k]; af.y = arow[k + 1];
    const float* vk = vb + (size_t)k * DD;
    b0.x = vk[0];       b0.y = vk[DD];
    b1.x = vk[16];      b1.y = vk[DD + 16];
    b2.x = vk[32];      b2.y = vk[DD + 32];
    b3.x = vk[48];      b3.y = vk[DD + 48];
    acc0 = wmma4(af, b0, acc0);
    acc1 = wmma4(af, b1, acc1);
    acc2 = wmma4(af, b2, acc2);
    acc3 = wmma4(af, b3, acc3);
  }

  #pragma unroll
  for (int r = 0; r < 8; ++r) {
    const int q = qt * 16 + r + 8 * hi;
    float* crow = ctx + ((size_t)(b * SS + q)) * DD + h * DKK + lr;
    crow[0]  = acc0[r];
    crow[16] = acc1[r];
    crow[32] = acc2[r];
    crow[48] = acc3[r];
  }
}

// ---------------------------------------------------------------------------
// Launch
// ---------------------------------------------------------------------------
extern "C" void kernel_launch(void* const* d_in, const int* in_sizes, int n_in,
                              void* d_out, int out_size, void* d_ws, size_t ws_size,
                              hipStream_t stream) {
  const float* x  = (const float*)d_in[0];
  // d_in[1] = mask (causal tril) — applied analytically
  const float* wq = (const float*)d_in[2];
  const float* bq = (const float*)d_in[3];
  const float* wk = (const float*)d_in[4];
  const float* bk = (const float*)d_in[5];
  const float* wv = (const float*)d_in[6];
  const float* bv = (const float*)d_in[7];
  const float* wo = (const float*)d_in[8];
  const float* bo = (const float*)d_in[9];
  const float* g  = (const float*)d_in[10];

  const size_t NSD = (size_t)BB * SS * DD;   // 4,194,304
  float* out1 = (float*)d_out;               // [B,S,D] residual output
  float* attn = out1 + NSD;                  // [B,H,S,S] attention output

  float* wsf    = (float*)d_ws;
  float* normed = wsf;                       // [B*S, D]
  float* Qb     = wsf + 1 * NSD;             // [B,S,D] (heads in columns)
  float* Kb     = wsf + 2 * NSD;
  float* Vb     = wsf + 3 * NSD;
  float* ctx    = wsf + 4 * NSD;             // [B,S,D]

  const int M = BB * SS;                     // 4096 rows

  // 1) RMSNorm
  rmsnorm_kernel<<<M, 256, 0, stream>>>(x, g, normed);

  // 2) QKV projections: macro-tiles = (M/32)*(D/64) = 2048 -> 256 blocks (8 waves)
  const int gemmBlocks = (M / 32) * (DD / 64) / 8;
  gemm_wt32x64_kernel<<<gemmBlocks, 256, 0, stream>>>(normed, wq, bq, nullptr, Qb, M, DD, DD);
  gemm_wt32x64_kernel<<<gemmBlocks, 256, 0, stream>>>(normed, wk, bk, nullptr, Kb, M, DD, DD);
  gemm_wt32x64_kernel<<<gemmBlocks, 256, 0, stream>>>(normed, wv, bv, nullptr, Vb, M, DD, DD);

  // 3) Scores (+causal mask): strips = B*H*128*32 = 131072 -> 16384 blocks
  const int scoreBlocks = BB * HH * (SS / 16) * (SS / 64) / 8;
  scores_kernel<<<scoreBlocks, 256, 0, stream>>>(Qb, Kb, attn);

  // 4) Softmax rows: B*H*S = 65536 rows
  softmax_kernel<<<BB * HH * SS, 256, 0, stream>>>(attn);

  // 5) attn @ V: strips = B*H*128 = 4096 -> 512 blocks
  const int pvBlocks = BB * HH * (SS / 16) / 8;
  pv_kernel<<<pvBlocks, 256, 0, stream>>>(attn, Vb, ctx);

  // 6) Output projection + residual: out1 = ctx @ Wo^T + bo + x
  gemm_wt32x64_kernel<<<gemmBlocks, 256, 0, stream>>>(ctx, wo, bo, x, out1, M, DD, DD);
}